// tdegnn_temporal_51445118271519
// MI455X (gfx1250) — compile-verified
//
#include <hip/hip_runtime.h>
#include <math.h>

typedef float v2f __attribute__((ext_vector_type(2)));
typedef float v8f __attribute__((ext_vector_type(8)));

#define NHID 128
#define NIN 12
#define NOUT 12
#define NLAYERS 4
#define BN_EPS 1e-5f
#define HCONST 0.1f

// scalar slots (floats) in workspace
#define SC_S0    0
#define SC_S1    1
#define SC_SNR   2
#define SC_ALPHA 3
#define SC_BETA  4
#define SC_NRO   5
#define SC_DONE  6
#define SC_KEEP  7
#define SC_ATT   8    // 16 floats
#define SC_C4    24   // 4 floats

// ---------------------------------------------------------------- reductions
__device__ __forceinline__ float block_sum_256(float v) {
  __shared__ float sh[256];
  __syncthreads();
  sh[threadIdx.x] = v;
  __syncthreads();
#pragma unroll
  for (int s = 128; s > 0; s >>= 1) {
    if ((int)threadIdx.x < s) sh[threadIdx.x] += sh[threadIdx.x + s];
    __syncthreads();
  }
  return sh[0];
}

// ---------------------------------------------------------------- WMMA GEMM
// Y[n, OUTW] = epilogue( concat(X0|X1|X2)[n, Ktot] @ W[OUTW, Ktot]^T + bias )
// mode 0: +bias; mode 1: +bias, batchnorm(bn = g,b,m,v rows of 128), relu;
// mode 2: +bias, clip(-1,1).
// One wave computes a 16-row x OUTW-col tile via v_wmma_f32_16x16x4_f32.
// k-loop unrolled by 2 with ping-pong fragment registers: no copy overhead,
// loads for step s+1 / s+2 overlap the WMMA burst for step s.
template <int OUTW>
__global__ __launch_bounds__(256) void mm_wmma(
    const float* __restrict__ X0, int k0,
    const float* __restrict__ X1, int k1,
    const float* __restrict__ X2, int k2,
    const float* __restrict__ W, const float* __restrict__ bias,
    const float* __restrict__ bn, float* __restrict__ Y,
    int n, int mode)
{
  constexpr int TILES = (OUTW + 15) / 16;
  const int lane = threadIdx.x & 31;
  const int wave = threadIdx.x >> 5;
  const int rowBase = blockIdx.x * 128 + wave * 16;
  if (rowBase >= n) return;                 // wave-uniform: EXEC stays full
  const int Ktot = k0 + k1 + k2;
  const int nsteps = Ktot >> 2;             // k-steps of 4
  const int mrow = lane & 15;
  const int ksel = (lane >> 4) << 1;        // lanes 0-15 -> K0/K1, 16-31 -> K2/K3

  int rowA = rowBase + mrow;
  if (rowA >= n) rowA = n - 1;              // clamp loads, mask stores later

  // k-invariant W row pointers per output tile (clamped when OUTW < 16*TILES)
  const float* wp[TILES];
#pragma unroll
  for (int t = 0; t < TILES; ++t) {
    int col  = t * 16 + mrow;
    int colc = col < OUTW ? col : OUTW - 1;
    wp[t] = W + (size_t)colc * Ktot;
    __builtin_prefetch(wp[t], 0, 0);        // global_prefetch_b8
  }

  v8f acc[TILES];
#pragma unroll
  for (int t = 0; t < TILES; ++t)
#pragma unroll
    for (int q = 0; q < 8; ++q) acc[t][q] = 0.f;

  auto loadFrag = [&](int step, v2f& a, v2f (&b)[TILES]) {
    const int kk = step << 2;
    const float* Xs; int kl, kd;
    if (kk < k0)            { Xs = X0; kl = kk;           kd = k0; }
    else if (kk < k0 + k1)  { Xs = X1; kl = kk - k0;      kd = k1; }
    else                    { Xs = X2; kl = kk - k0 - k1; kd = k2; }
    const float* p = Xs + (size_t)rowA * kd + kl + ksel;
    a.x = p[0]; a.y = p[1];
#pragma unroll
    for (int t = 0; t < TILES; ++t) {
      b[t].x = wp[t][kk + ksel];
      b[t].y = wp[t][kk + ksel + 1];
    }
  };

  auto domm = [&](const v2f& a, const v2f (&b)[TILES]) {
#pragma unroll
    for (int t = 0; t < TILES; ++t)
      acc[t] = __builtin_amdgcn_wmma_f32_16x16x4_f32(
          false, a, false, b[t], (short)0, acc[t], false, false);
  };

  v2f aP, aQ;
  v2f bP[TILES], bQ[TILES];
  loadFrag(0, aP, bP);                      // prologue: step 0 -> P

  int s = 0;
  for (; s + 2 <= nsteps; s += 2) {
    loadFrag(s + 1, aQ, bQ);                // prefetch step s+1 -> Q
    domm(aP, bP);                           // compute step s
    int sn = s + 2 < nsteps ? s + 2 : nsteps - 1;  // clamped (dup load, unused)
    loadFrag(sn, aP, bP);                   // prefetch step s+2 -> P
    domm(aQ, bQ);                           // compute step s+1
  }
  if (s < nsteps) domm(aP, bP);             // odd tail (nsteps odd)

  const int rowOff = (lane >> 4) << 3;      // 0 or 8
#pragma unroll
  for (int t = 0; t < TILES; ++t) {
    int col = t * 16 + mrow;
    if (col >= OUTW) continue;
    float bc = bias[col];
#pragma unroll
    for (int q = 0; q < 8; ++q) {
      int row = rowBase + rowOff + q;
      if (row < n) {
        float x = acc[t][q] + bc;
        if (mode == 1) {
          x = (x - bn[2 * NHID + col]) * rsqrtf(bn[3 * NHID + col] + BN_EPS)
              * bn[col] + bn[NHID + col];
          x = fmaxf(x, 0.f);
        } else if (mode == 2) {
          x = fminf(fmaxf(x, -1.f), 1.f);
        }
        Y[(size_t)row * OUTW + col] = x;
      }
    }
  }
}

// ---------------------------------------------------------------- small kernels
__global__ void k_zero_sc(float* sc) {
  if (threadIdx.x < 32) sc[threadIdx.x] = 0.f;
}
__global__ void k_zero16(float* p) { if (threadIdx.x < 16) p[threadIdx.x] = 0.f; }

__global__ void k_deg_init(float* deg, int n) {
  int i = blockIdx.x * blockDim.x + threadIdx.x;
  if (i < n) deg[i] = 1.f;                       // self loop
}
__global__ void k_deg_count(const int* __restrict__ ei, int E, float* deg) {
  int e = blockIdx.x * blockDim.x + threadIdx.x;
  if (e < E) atomicAdd(&deg[ei[e]], 1.f);        // row = src
}
__global__ void k_dinv(const float* __restrict__ deg, float* dnv, int n) {
  int i = blockIdx.x * blockDim.x + threadIdx.x;
  if (i < n) dnv[i] = deg[i] > 0.f ? rsqrtf(deg[i]) : 0.f;
}

// te[n,l] = silu( sum_c te_w[c]*tf[n,c,l] + te_b )
__global__ void k_te(const float* __restrict__ tf, const float* __restrict__ tw,
                     const float* __restrict__ tb, float* __restrict__ te, int n) {
  int i = blockIdx.x * blockDim.x + threadIdx.x;
  if (i >= n * NIN) return;
  int nn = i / NIN, l = i % NIN;
  float s = tb[0];
#pragma unroll
  for (int c = 0; c < 20; ++c) s += tw[c] * tf[(size_t)nn * 240 + c * NIN + l];
  te[i] = s * (1.f / (1.f + expf(-s)));
}

// emb slice l: relu(bn_state(conv_w[c]*T[n,l]+conv_b[c]))
__global__ void k_emb(const float* __restrict__ T, const float* __restrict__ cw,
                      const float* __restrict__ cb, const float* __restrict__ bn,
                      float* __restrict__ out, int n, int l) {
  int i = blockIdx.x * blockDim.x + threadIdx.x;
  if (i >= n * NHID) return;
  int nn = i >> 7, c = i & 127;
  float x = cw[c] * T[(size_t)nn * NIN + l] + cb[c];
  x = (x - bn[2 * NHID + c]) * rsqrtf(bn[3 * NHID + c] + BN_EPS) * bn[c] + bn[NHID + c];
  out[i] = fmaxf(x, 0.f);
}

// per-node 4x4 Gram -> softmax -> log(p*relu(f)+1e-4) -> global 16-slot sum
__global__ __launch_bounds__(256) void k_attn(
    const float* __restrict__ Q, const float* __restrict__ K, size_t stride,
    const float* __restrict__ mha, float* __restrict__ acc16, int n)
{
  __shared__ float sh[16];
  if (threadIdx.x < 16) sh[threadIdx.x] = 0.f;
  __syncthreads();
  int i = blockIdx.x * 256 + threadIdx.x;
  if (i < n) {
    float dot[16];
#pragma unroll
    for (int t = 0; t < 16; ++t) dot[t] = 0.f;
    const float* qb = Q + (size_t)i * NHID;
    const float* kb = K + (size_t)i * NHID;
    for (int e = 0; e < NHID; ++e) {
      float qv[4], kv[4];
#pragma unroll
      for (int l = 0; l < 4; ++l) { qv[l] = qb[l * stride + e]; kv[l] = kb[l * stride + e]; }
#pragma unroll
      for (int l = 0; l < 4; ++l)
#pragma unroll
        for (int k = 0; k < 4; ++k) dot[l * 4 + k] += qv[l] * kv[k];
    }
    float f = fmaxf(mha[0], 0.f);
    const float scale = 0.0883883476483184f;  // 1/sqrt(128)
    for (int l = 0; l < 4; ++l) {
      float m = -3.4e38f;
      for (int k = 0; k < 4; ++k) m = fmaxf(m, dot[l * 4 + k] * scale);
      float ex[4]; float s = 0.f;
      for (int k = 0; k < 4; ++k) { ex[k] = expf(dot[l * 4 + k] * scale - m); s += ex[k]; }
      for (int k = 0; k < 4; ++k)
        atomicAdd(&sh[l * 4 + k], logf((ex[k] / s) * f + 1e-4f));
    }
  }
  __syncthreads();
  if (threadIdx.x < 16) atomicAdd(&acc16[threadIdx.x], sh[threadIdx.x]);
}

__global__ void k_attn_fin(float* sc, float invn) {
  float A[16];
  for (int i = 0; i < 16; ++i) A[i] = sc[SC_ATT + i] * invn;  // mean over n
  float c[4]; float s = 0.f;
  for (int k = 0; k < 4; ++k) { c[k] = 0.5f * (A[12 + k] + A[k * 4 + 3]); s += c[k]; }
  for (int k = 0; k < 4; ++k) sc[SC_C4 + k] = c[k] / s;
}

__global__ void k_ts1(const float* __restrict__ a0, const float* __restrict__ a1,
                      const float* __restrict__ a2, const float* __restrict__ a3,
                      const float* __restrict__ sc, float* __restrict__ out, int total) {
  int i = blockIdx.x * blockDim.x + threadIdx.x;
  if (i >= total) return;
  out[i] = sc[SC_C4] * a0[i] + sc[SC_C4 + 1] * a1[i]
         + sc[SC_C4 + 2] * a2[i] + sc[SC_C4 + 3] * a3[i];
}

// Tstate = relu(bn_react(Tstate1 + H*(m1 + m0 + Tstate1*m2clip)))
__global__ void k_react(const float* __restrict__ t1, const float* __restrict__ m1,
                        const float* __restrict__ m0, const float* __restrict__ m2,
                        const float* __restrict__ bn, float* __restrict__ X, int total) {
  int i = blockIdx.x * blockDim.x + threadIdx.x;
  if (i >= total) return;
  int c = i & 127;
  float x = t1[i] + HCONST * (m1[i] + m0[i] + t1[i] * m2[i]);
  x = (x - bn[2 * NHID + c]) * rsqrtf(bn[3 * NHID + c] + BN_EPS) * bn[c] + bn[NHID + c];
  X[i] = fmaxf(x, 0.f);
}

// OUT = (1 + H*Kd[c]) * Y  (diagonal part of matvec; scatter adds the rest)
__global__ void k_mv_init(const float* __restrict__ Y, const float* __restrict__ kap,
                          float* __restrict__ OUT, int total) {
  int i = blockIdx.x * blockDim.x + threadIdx.x;
  if (i >= total) return;
  int c = i & 127;
  float kd = fminf(fmaxf(kap[c], 0.f), 1.f);
  OUT[i] = (1.f + HCONST * kd) * Y[i];
}

// edge (plus self-loop) scatter: OUT[row,c] += H*Kd[c] * wlap * Y[col,c]
__global__ __launch_bounds__(128) void k_scatter(
    const int* __restrict__ ei, int E, int n, const float* __restrict__ dnv,
    const float* __restrict__ kap, const float* __restrict__ Y, float* __restrict__ OUT)
{
  int e = blockIdx.x;
  int c = threadIdx.x;
  int r, cl;
  if (e < E) { r = ei[e]; cl = ei[E + e]; } else { r = e - E; cl = r; }
  float wgt = -dnv[r] * dnv[cl];
  float kd = fminf(fmaxf(kap[c], 0.f), 1.f);
  atomicAdd(&OUT[(size_t)r * NHID + c], HCONST * kd * wgt * Y[(size_t)cl * NHID + c]);
}

__global__ __launch_bounds__(256) void k_rinit(
    const float* __restrict__ Ts, const float* __restrict__ LP,
    float* __restrict__ R, float* __restrict__ P, float* sc, int total)
{
  int i = blockIdx.x * 256 + threadIdx.x;
  float v = 0.f;
  if (i < total) { float r = Ts[i] - LP[i]; R[i] = r; P[i] = r; v = r * r; }
  float s = block_sum_256(v);
  if (threadIdx.x == 0) atomicAdd(&sc[SC_S0], s);
}

__global__ void k_cg_start(float* sc, float NC) {
  sc[SC_NRO] = sc[SC_S0] / NC;
  sc[SC_DONE] = 0.f; sc[SC_KEEP] = 0.f;
  sc[SC_S0] = 0.f; sc[SC_S1] = 0.f; sc[SC_SNR] = 0.f;
}

__global__ __launch_bounds__(256) void k_reduce2(
    const float* __restrict__ A, const float* __restrict__ B,
    const float* __restrict__ C, const float* __restrict__ D, float* sc, int total)
{
  int i = blockIdx.x * 256 + threadIdx.x;
  float v0 = 0.f, v1 = 0.f;
  if (i < total) { v0 = A[i] * B[i]; v1 = C[i] * D[i]; }
  float s0 = block_sum_256(v0);
  float s1 = block_sum_256(v1);
  if (threadIdx.x == 0) { atomicAdd(&sc[SC_S0], s0); atomicAdd(&sc[SC_S1], s1); }
}

__global__ void k_cg_mid(float* sc, float NC) {
  sc[SC_ALPHA] = (sc[SC_S0] / NC) / (sc[SC_S1] / NC + 1e-6f);
  sc[SC_SNR] = 0.f;
}

__global__ __launch_bounds__(256) void k_cg_update(
    float* __restrict__ X, float* __restrict__ R,
    const float* __restrict__ P, const float* __restrict__ LP, float* sc, int total)
{
  float alpha = sc[SC_ALPHA];
  float done = sc[SC_DONE];
  int i = blockIdx.x * 256 + threadIdx.x;
  float v = 0.f;
  if (i < total) {
    float r = R[i];
    float rn = (done != 0.f) ? r : (r - alpha * LP[i]);
    if (done == 0.f) { X[i] += alpha * P[i]; R[i] = rn; }
    v = rn * rn;
  }
  float s = block_sum_256(v);
  if (threadIdx.x == 0) atomicAdd(&sc[SC_SNR], s);
}

__global__ void k_cg_post(float* sc, float NC) {
  float nr = sc[SC_SNR] / NC;
  float done = sc[SC_DONE];
  float nro = sc[SC_NRO];
  int keep = (done != 0.f) || (nr < 1e-5f);
  sc[SC_BETA] = nr / (nro + 1e-6f);
  if (!keep) sc[SC_NRO] = nr;
  sc[SC_KEEP] = keep ? 1.f : 0.f;
  sc[SC_DONE] = keep ? 1.f : 0.f;
  sc[SC_S0] = 0.f; sc[SC_S1] = 0.f;
}

__global__ void k_cg_pupd(float* __restrict__ P, const float* __restrict__ R,
                          const float* sc, int total) {
  float keep = sc[SC_KEEP];
  float beta = sc[SC_BETA];
  int i = blockIdx.x * blockDim.x + threadIdx.x;
  if (keep == 0.f && i < total) P[i] = R[i] + beta * P[i];
}

__global__ void k_lsm(const float* __restrict__ Z, float* __restrict__ out, int n) {
  int i = blockIdx.x * blockDim.x + threadIdx.x;
  if (i >= n) return;
  float m = -3.4e38f;
#pragma unroll
  for (int k = 0; k < NOUT; ++k) m = fmaxf(m, Z[(size_t)i * NOUT + k]);
  float s = 0.f;
#pragma unroll
  for (int k = 0; k < NOUT; ++k) s += expf(Z[(size_t)i * NOUT + k] - m);
  float ls = logf(s);
#pragma unroll
  for (int k = 0; k < NOUT; ++k)
    out[(size_t)i * NOUT + k] = Z[(size_t)i * NOUT + k] - m - ls;
}

// ---------------------------------------------------------------- driver
extern "C" void kernel_launch(void* const* d_in, const int* in_sizes, int n_in,
                              void* d_out, int out_size, void* d_ws, size_t ws_size,
                              hipStream_t stream)
{
  const float* T        = (const float*)d_in[0];
  const float* TF       = (const float*)d_in[1];
  const int*   EI       = (const int*)d_in[2];
  const float* Wopen    = (const float*)d_in[3];
  const float* bopen    = (const float*)d_in[4];
  const float* bnOpenH  = (const float*)d_in[5];
  const float* bnOpenS  = (const float*)d_in[6];
  const float* convW    = (const float*)d_in[7];
  const float* convB    = (const float*)d_in[8];
  const float* teW      = (const float*)d_in[9];
  const float* teB      = (const float*)d_in[10];
  const float* KR1w     = (const float*)d_in[11];
  const float* KR1b     = (const float*)d_in[12];
  const float* KR2w     = (const float*)d_in[13];
  const float* KR2b     = (const float*)d_in[14];
  const float* KRU0w    = (const float*)d_in[15];
  const float* KRU0b    = (const float*)d_in[16];
  const float* kappa    = (const float*)d_in[17];
  const float* Hew      = (const float*)d_in[18];
  const float* Heb      = (const float*)d_in[19];
  const float* rsw      = (const float*)d_in[20];
  const float* rsb      = (const float*)d_in[21];
  const float* bnReact  = (const float*)d_in[22];
  const float* bnHist   = (const float*)d_in[23];
  const float* bnResc   = (const float*)d_in[24];
  const float* Wq       = (const float*)d_in[25];
  const float* bq       = (const float*)d_in[26];
  const float* Wk       = (const float*)d_in[27];
  const float* bk       = (const float*)d_in[28];
  const float* mha      = (const float*)d_in[29];
  const float* Wclose   = (const float*)d_in[30];
  const float* bclose   = (const float*)d_in[31];
  (void)n_in; (void)out_size; (void)ws_size;

  const int n = in_sizes[0] / NIN;
  const int E = in_sizes[2] / 2;
  const size_t NB = (size_t)n * NHID;
  const int totNB = (int)NB;

  // workspace carve-up
  float* w = (float*)d_ws;
  float* e0 = w; w += NB; float* e1 = w; w += NB;
  float* e2 = w; w += NB; float* e3 = w; w += NB;
  float* L[4]; for (int j = 0; j < 4; ++j) { L[j] = w; w += NB; }
  float* t0  = w; w += NB;
  float* thA = w; w += NB;
  float* thB = w; w += NB;
  float* tsR = w; w += NB;
  float* S   = w; w += 8 * NB;        // s0..s7 contiguous (Q0..3, K0..3 / CG temps)
  float* te  = w; w += (size_t)n * NIN;
  float* deg = w; w += n;
  float* dnv = w; w += n;
  float* sc  = w; w += 64;

  dim3 B256(256);
  const int gElem = (totNB + 255) / 256;
  const int gRow  = (n + 255) / 256;
  const int gMM   = (n + 127) / 128;
  const float NC  = (float)NB;

  k_zero_sc<<<1, 32, 0, stream>>>(sc);

  // graph degree / dinv
  k_deg_init<<<gRow, B256, 0, stream>>>(deg, n);
  k_deg_count<<<(E + 255) / 256, B256, 0, stream>>>(EI, E, deg);
  k_dinv<<<gRow, B256, 0, stream>>>(deg, dnv, n);

  // open: Thist = relu(bn(T @ Wopen^T + bopen)); T0 = copy
  mm_wmma<NHID><<<gMM, B256, 0, stream>>>(T, NIN, nullptr, 0, nullptr, 0,
                                          Wopen, bopen, bnOpenH, thA, n, 1);
  hipMemcpyAsync(t0, thA, NB * sizeof(float), hipMemcpyDeviceToDevice, stream);

  // acts = emb[0..3]; initial Tstate = emb[11] -> s7
  k_emb<<<gElem, B256, 0, stream>>>(T, convW, convB, bnOpenS, e0, n, 0);
  k_emb<<<gElem, B256, 0, stream>>>(T, convW, convB, bnOpenS, e1, n, 1);
  k_emb<<<gElem, B256, 0, stream>>>(T, convW, convB, bnOpenS, e2, n, 2);
  k_emb<<<gElem, B256, 0, stream>>>(T, convW, convB, bnOpenS, e3, n, 3);
  k_emb<<<gElem, B256, 0, stream>>>(T, convW, convB, bnOpenS, S + 7 * NB, n, NIN - 1);

  // time embedding
  k_te<<<(n * NIN + 255) / 256, B256, 0, stream>>>(TF, teW, teB, te, n);

  float* acts[4] = { e0, e1, e2, e3 };
  float* thCur = thA; float* thNxt = thB;
  float* Ts = S + 7 * NB;

  for (int j = 0; j < NLAYERS; ++j) {
    const float* kap = kappa + (size_t)j * NHID;

    // 1. Tstate = relu(bn(concat(te,Tstate) @ rs^T))
    mm_wmma<NHID><<<gMM, B256, 0, stream>>>(te, NIN, Ts, NHID, nullptr, 0,
        rsw + (size_t)j * NHID * (NIN + NHID), rsb + (size_t)j * NHID,
        bnResc + (size_t)j * 4 * NHID, tsR, n, 1);

    // 2. Thist = relu(bn(concat(T0,Thist,Tstate) @ He^T))
    mm_wmma<NHID><<<gMM, B256, 0, stream>>>(t0, NHID, thCur, NHID, tsR, NHID,
        Hew + (size_t)j * NHID * 3 * NHID, Heb + (size_t)j * NHID,
        bnHist + (size_t)j * 4 * NHID, thNxt, n, 1);
    { float* tmp = thCur; thCur = thNxt; thNxt = tmp; }

    // 3. Q,K projections of the 4 act stacks -> S[0..3], S[4..7]
    for (int l = 0; l < 4; ++l) {
      mm_wmma<NHID><<<gMM, B256, 0, stream>>>(acts[l], NHID, nullptr, 0, nullptr, 0,
          Wq, bq, nullptr, S + (size_t)l * NB, n, 0);
      mm_wmma<NHID><<<gMM, B256, 0, stream>>>(acts[l], NHID, nullptr, 0, nullptr, 0,
          Wk, bk, nullptr, S + (size_t)(4 + l) * NB, n, 0);
    }

    // 4. attention statistic -> C4
    k_zero16<<<1, 16, 0, stream>>>(sc + SC_ATT);
    k_attn<<<gRow, B256, 0, stream>>>(S, S + 4 * NB, NB, mha, sc + SC_ATT, n);
    k_attn_fin<<<1, 1, 0, stream>>>(sc, 1.f / (float)n);

    // 5. Tstate1 = sum_l C[l]*acts[l] -> s7
    k_ts1<<<gElem, B256, 0, stream>>>(acts[0], acts[1], acts[2], acts[3],
                                      sc, S + 7 * NB, totNB);

    // 6. dT matmuls -> s1,s2,s3
    mm_wmma<NHID><<<gMM, B256, 0, stream>>>(thCur, NHID, nullptr, 0, nullptr, 0,
        KR1w + (size_t)j * NHID * NHID, KR1b + (size_t)j * NHID, nullptr,
        S + 1 * NB, n, 0);
    mm_wmma<NHID><<<gMM, B256, 0, stream>>>(t0, NHID, nullptr, 0, nullptr, 0,
        KRU0w + (size_t)j * NHID * NHID, KRU0b + (size_t)j * NHID, nullptr,
        S + 2 * NB, n, 0);
    mm_wmma<NHID><<<gMM, B256, 0, stream>>>(thCur, NHID, nullptr, 0, nullptr, 0,
        KR2w + (size_t)j * NHID * NHID, KR2b + (size_t)j * NHID, nullptr,
        S + 3 * NB, n, 2);

    // 7. react -> L[j] (CG's X, in place)
    k_react<<<gElem, B256, 0, stream>>>(S + 7 * NB, S + 1 * NB, S + 2 * NB,
                                        S + 3 * NB, bnReact + (size_t)j * 4 * NHID,
                                        L[j], totNB);

    // 8. CG solve (I + H*Kd*Lap) X = Tstate, 5 branchless iterations
    float* X  = L[j];
    float* R  = S + 4 * NB;
    float* P  = S + 5 * NB;
    float* LP = S + 6 * NB;

    k_mv_init<<<gElem, B256, 0, stream>>>(X, kap, LP, totNB);
    k_scatter<<<dim3(E + n), dim3(128), 0, stream>>>(EI, E, n, dnv, kap, X, LP);
    k_rinit<<<gElem, B256, 0, stream>>>(X, LP, R, P, sc, totNB);
    k_cg_start<<<1, 1, 0, stream>>>(sc, NC);

    for (int it = 0; it < 5; ++it) {
      k_mv_init<<<gElem, B256, 0, stream>>>(P, kap, LP, totNB);
      k_scatter<<<dim3(E + n), dim3(128), 0, stream>>>(EI, E, n, dnv, kap, P, LP);
      k_reduce2<<<gElem, B256, 0, stream>>>(R, R, P, LP, sc, totNB);
      k_cg_mid<<<1, 1, 0, stream>>>(sc, NC);
      k_cg_update<<<gElem, B256, 0, stream>>>(X, R, P, LP, sc, totNB);
      k_cg_post<<<1, 1, 0, stream>>>(sc, NC);
      k_cg_pupd<<<gElem, B256, 0, stream>>>(P, R, sc, totNB);
    }

    // 9. rotate acts; Tstate = X
    acts[0] = acts[1]; acts[1] = acts[2]; acts[2] = acts[3]; acts[3] = X;
    Ts = X;
  }

  // close + log_softmax
  mm_wmma<NOUT><<<gMM, B256, 0, stream>>>(Ts, NHID, nullptr, 0, nullptr, 0,
                                          Wclose, bclose, nullptr, S, n, 0);
  k_lsm<<<gRow, B256, 0, stream>>>(S, (float*)d_out, n);
}